// SparseConv2d_64312840290399
// MI455X (gfx1250) — compile-verified
//
#include <hip/hip_runtime.h>

#define IC   256
#define OC   256
#define HH   56
#define WW   56
#define HW   (HH * WW)
#define NB   32
#define NP   (NB * HW)      // 100352 output pixels
#define TAPS 9
#define PXT  64             // pixels per block (64 | 3136)
#define LSTR 132            // LDS row stride in floats (132 % 64 == 4 -> conflict-free b64)
#define KCH  128            // ic chunk staged per LDS fill

typedef __attribute__((ext_vector_type(2))) float v2f;
typedef __attribute__((ext_vector_type(8))) float v8f;

// ---------------------------------------------------------------------------
// Prep: fold sparsity mask into weights and transpose OIHW -> [tap][oc][ic]
// so WMMA A-fragments are contiguous float2 loads per lane.
// ---------------------------------------------------------------------------
__global__ __launch_bounds__(256) void prep_weights(
    const float* __restrict__ w, const int* __restrict__ m,
    float* __restrict__ wp) {
  int e = blockIdx.x * 256 + threadIdx.x;     // over OC*IC*9 (OIHW flat)
  if (e >= OC * IC * TAPS) return;
  int oc = e / (IC * TAPS);
  int r  = e - oc * (IC * TAPS);
  int ic = r / TAPS;
  int t  = r - ic * TAPS;                     // t = kh*3 + kw
  float v = m[e] ? w[e] : 0.0f;
  wp[(size_t)t * OC * IC + (size_t)oc * IC + ic] = v;
}

// ---------------------------------------------------------------------------
// Implicit-GEMM conv via V_WMMA_F32_16X16X4_F32.
// Block = 256 threads (8 wave32): 64 pixels x 256 OC.
// Per wave: 32 OC x 64 pixels = 2 A-frags x 4 B-frags = 8 accumulators.
// K-dim = 9 taps x 256 IC; each tap staged into LDS in two 128-ic chunks,
// LDS layout [pixel][ic] (stride 132) so B-frags are single ds_load_b64.
// ---------------------------------------------------------------------------
__global__ __launch_bounds__(256) void sparse_conv_wmma(
    const float* __restrict__ in, const float* __restrict__ wp,
    const float* __restrict__ bias, float* __restrict__ out) {
  __shared__ float sB[PXT * LSTR];            // 64*132*4 = 33792 B

  const int tid  = threadIdx.x;
  const int wid  = tid >> 5;
  const int lane = tid & 31;
  const int c    = lane & 15;                 // N-column within 16
  const int koff = (lane >> 4) << 1;          // 0 or 2 (A/B K-sublane split)

  const int p0   = blockIdx.x * PXT;          // 64 | 3136 -> same image
  const int n    = p0 / HW;
  const int rem0 = p0 - n * HW;

  // staging role: thread owns pixel (tid&63), channels {tid>>6 + 4j}
  const int spix = tid & 63;
  const int sic0 = tid >> 6;                  // 0..3
  const int srem = rem0 + spix;
  const int soh  = srem / WW;
  const int sow  = srem - soh * WW;
  const float* inN  = in + (size_t)n * IC * HW;
  float*       sRow = sB + spix * LSTR;

  const int oc0 = wid * 32;
  v8f acc[2][4] = {};

  for (int tap = 0; tap < TAPS; ++tap) {
    const int dh = tap / 3 - 1;
    const int dw = tap - (tap / 3) * 3 - 1;
    const int ih = soh + dh;
    const int iw = sow + dw;
    const bool valid = (ih >= 0) && (ih < HH) && (iw >= 0) && (iw < WW);
    const float* gsrc = inN + ih * WW + iw;

    const float* wt     = wp + (size_t)tap * OC * IC;
    const float* a0base = wt + (size_t)(oc0 + c) * IC + koff;
    const float* a1base = wt + (size_t)(oc0 + 16 + c) * IC + koff;

    for (int h = 0; h < 2; ++h) {             // two 128-ic chunks per tap
      __syncthreads();                        // previous chunk's reads done
#pragma unroll 8
      for (int j = 0; j < 32; ++j) {
        const int jc = j * 4 + sic0;          // 0..127
        float v = 0.0f;
        if (valid) v = gsrc[(h * KCH + jc) * HW];
        sRow[jc] = v;
      }
      __syncthreads();                        // chunk visible to all waves

      const float* a0b = a0base + h * KCH;
      const float* a1b = a1base + h * KCH;

#pragma unroll 4
      for (int kb = 0; kb < KCH; kb += 4) {
        v2f A0 = *(const v2f*)(a0b + kb);     // A[m=c][k..k+1], 8B aligned
        v2f A1 = *(const v2f*)(a1b + kb);
        const float* bp = sB + c * LSTR + koff + kb;
        v2f B0 = *(const v2f*)(bp);           // pixel tiles 0..3
        v2f B1 = *(const v2f*)(bp + 16 * LSTR);
        v2f B2 = *(const v2f*)(bp + 32 * LSTR);
        v2f B3 = *(const v2f*)(bp + 48 * LSTR);
        acc[0][0] = __builtin_amdgcn_wmma_f32_16x16x4_f32(
            false, A0, false, B0, (short)0, acc[0][0], false, false);
        acc[0][1] = __builtin_amdgcn_wmma_f32_16x16x4_f32(
            false, A0, false, B1, (short)0, acc[0][1], false, false);
        acc[0][2] = __builtin_amdgcn_wmma_f32_16x16x4_f32(
            false, A0, false, B2, (short)0, acc[0][2], false, false);
        acc[0][3] = __builtin_amdgcn_wmma_f32_16x16x4_f32(
            false, A0, false, B3, (short)0, acc[0][3], false, false);
        acc[1][0] = __builtin_amdgcn_wmma_f32_16x16x4_f32(
            false, A1, false, B0, (short)0, acc[1][0], false, false);
        acc[1][1] = __builtin_amdgcn_wmma_f32_16x16x4_f32(
            false, A1, false, B1, (short)0, acc[1][1], false, false);
        acc[1][2] = __builtin_amdgcn_wmma_f32_16x16x4_f32(
            false, A1, false, B2, (short)0, acc[1][2], false, false);
        acc[1][3] = __builtin_amdgcn_wmma_f32_16x16x4_f32(
            false, A1, false, B3, (short)0, acc[1][3], false, false);
      }
    }
  }

  // Epilogue: C/D layout -> VGPR r, lanes 0-15: M=r, lanes 16-31: M=r+8.
  const int mAdd = (lane >> 4) * 8;
  float* outP = out + (size_t)n * OC * HW + rem0;
#pragma unroll
  for (int a = 0; a < 2; ++a) {
#pragma unroll
    for (int r = 0; r < 8; ++r) {
      const int oc = oc0 + a * 16 + r + mAdd;
      const float bv = bias[oc];
      float* row = outP + (size_t)oc * HW;
#pragma unroll
      for (int pt = 0; pt < 4; ++pt) {
        row[pt * 16 + c] = acc[a][pt][r] + bv;
      }
    }
  }
}

// ---------------------------------------------------------------------------
extern "C" void kernel_launch(void* const* d_in, const int* in_sizes, int n_in,
                              void* d_out, int out_size, void* d_ws, size_t ws_size,
                              hipStream_t stream) {
  const float* in   = (const float*)d_in[0];  // (32,256,56,56) f32
  const float* w    = (const float*)d_in[1];  // (256,256,3,3)  f32
  const float* b    = (const float*)d_in[2];  // (256,)         f32
  const int*   mask = (const int*)d_in[3];    // (256,256,3,3)  bool->int

  float* wp  = (float*)d_ws;                  // 9*256*256*4 = 2.36 MB scratch
  float* out = (float*)d_out;

  prep_weights<<<(OC * IC * TAPS + 255) / 256, 256, 0, stream>>>(w, mask, wp);
  sparse_conv_wmma<<<NP / PXT, 256, 0, stream>>>(in, wp, b, out);
}